// Encoder_6141803233543
// MI455X (gfx1250) — compile-verified
//
#include <hip/hip_runtime.h>

typedef __attribute__((ext_vector_type(2))) float v2f;
typedef __attribute__((ext_vector_type(8))) float v8f;

#define THREADS 256

// ---------------- CSR build: degree count ----------------
__global__ void k_zero_u32(unsigned* __restrict__ p, int n) {
  int i = blockIdx.x * blockDim.x + threadIdx.x;
  if (i < n) p[i] = 0u;
}

__global__ void k_count(const long long* __restrict__ dst, unsigned* __restrict__ cnt, int ne) {
  int i = blockIdx.x * blockDim.x + threadIdx.x;
  if (i < ne) atomicAdd(&cnt[(int)dst[i]], 1u);   // native global_atomic_add_u32
}

__global__ void k_dinv(const unsigned* __restrict__ cnt, float* __restrict__ dinv, int n) {
  int i = blockIdx.x * blockDim.x + threadIdx.x;
  if (i < n) dinv[i] = rsqrtf((float)cnt[i] + 1.0f);  // +1 self-loop; deg>=1 always
}

// ---------------- hierarchical inclusive scan (u32) ----------------
__global__ void k_scan1(const unsigned* __restrict__ in, unsigned* __restrict__ out,
                        unsigned* __restrict__ bsum, int n) {
  __shared__ unsigned s[256];
  int i = blockIdx.x * 256 + threadIdx.x;
  s[threadIdx.x] = (i < n) ? in[i] : 0u;
  __syncthreads();
  #pragma unroll
  for (int off = 1; off < 256; off <<= 1) {
    unsigned t = (threadIdx.x >= off) ? s[threadIdx.x - off] : 0u;
    __syncthreads();
    s[threadIdx.x] += t;
    __syncthreads();
  }
  if (i < n) out[i] = s[threadIdx.x];
  if (threadIdx.x == 255) bsum[blockIdx.x] = s[255];
}

__global__ void k_scan2(unsigned* __restrict__ bsum, int nb) {  // nb <= 512, one block of 512
  __shared__ unsigned s[512];
  s[threadIdx.x] = (threadIdx.x < (unsigned)nb) ? bsum[threadIdx.x] : 0u;
  __syncthreads();
  #pragma unroll
  for (int off = 1; off < 512; off <<= 1) {
    unsigned t = (threadIdx.x >= off) ? s[threadIdx.x - off] : 0u;
    __syncthreads();
    s[threadIdx.x] += t;
    __syncthreads();
  }
  if (threadIdx.x < (unsigned)nb) bsum[threadIdx.x] = s[threadIdx.x];
}

__global__ void k_scan3(unsigned* __restrict__ out, const unsigned* __restrict__ bsum, int n) {
  int i = blockIdx.x * 256 + threadIdx.x;
  if (i < n && blockIdx.x > 0) out[i] += bsum[blockIdx.x - 1];
}

__global__ void k_cursor(const unsigned* __restrict__ rs, unsigned* __restrict__ cur, int n) {
  int i = blockIdx.x * blockDim.x + threadIdx.x;
  if (i < n) cur[i] = (i == 0) ? 0u : rs[i - 1];   // exclusive start
}

__global__ void k_fill_edges(const long long* __restrict__ ei, unsigned* __restrict__ cur,
                             unsigned* __restrict__ esrc, int ne) {
  int e = blockIdx.x * blockDim.x + threadIdx.x;
  if (e >= ne) return;
  int s = (int)ei[e];
  int d = (int)ei[(size_t)ne + e];
  unsigned pos = atomicAdd(&cur[d], 1u);
  esrc[pos] = (unsigned)s;
}

// ---------------- dense projection: H[n,128] = X[n,128] @ W[128,128] ----------------
// 256 threads = 8 waves; wave computes a 16-row panel, 8 col tiles, K steps of 4.
// W staged in LDS pair-interleaved: sB[kp*128 + n] = {W[2kp][n], W[2kp+1][n]} so each
// B fragment is one aligned ds_load_b64 feeding v_wmma directly (no repack movs).
__global__ void k_gemm128_wmma(const float* __restrict__ X, const float* __restrict__ Wm,
                               float* __restrict__ H, int nrows) {
  extern __shared__ float smem[];          // 64 KB
  v2f* sB = (v2f*)smem;                    // [64 kp][128 n]
  const int tid = threadIdx.x;
  {
    #pragma unroll
    for (int i = tid; i < 64 * 128; i += THREADS) {
      int kp = i >> 7, n = i & 127;
      v2f w;
      w.x = Wm[(2 * kp    ) * 128 + n];
      w.y = Wm[(2 * kp + 1) * 128 + n];
      sB[i] = w;
    }
  }
  __syncthreads();

  const int wave = tid >> 5;
  const int lane = tid & 31;
  const int m = lane & 15;   // A row / B,D column within tile
  const int g = lane >> 4;   // half-wave select
  const int rowbase = blockIdx.x * 128 + wave * 16;

  v8f acc[8] = {};

  int arow = rowbase + m;
  if (arow >= nrows) arow = nrows - 1;          // clamp; OOB rows masked at store
  const v2f* xrow2 = (const v2f*)(X + (size_t)arow * 128);

  for (int kk = 0; kk < 32; ++kk) {
    const int kp = kk * 2 + g;                  // K pair handled by this half-wave
    v2f a = xrow2[kp];                          // A[m][2kp], A[m][2kp+1]
    const v2f* brow = sB + kp * 128 + m;
    #pragma unroll
    for (int t = 0; t < 8; ++t) {
      v2f b = brow[t * 16];                     // {B[2kp][n], B[2kp+1][n]}
      acc[t] = __builtin_amdgcn_wmma_f32_16x16x4_f32(
          false, a, false, b, (short)0, acc[t], false, false);
    }
  }

  #pragma unroll
  for (int t = 0; t < 8; ++t) {
    #pragma unroll
    for (int v = 0; v < 8; ++v) {
      int row = rowbase + v + 8 * g;
      if (row < nrows) H[(size_t)row * 128 + t * 16 + m] = acc[t][v];
    }
  }
}

// ---------------- CSR aggregation, fused self-loop + bias + ReLU ----------------
// one wave per node; lane owns 4 features (float4); gathers h[src] from L2.
__global__ void k_aggregate(const unsigned* __restrict__ rs,    // inclusive scan of counts
                            const unsigned* __restrict__ esrc,  // src id per CSR slot
                            const float* __restrict__ dinv,
                            const float* __restrict__ h,
                            const float* __restrict__ bias,
                            float* __restrict__ out, int n) {
  int node = blockIdx.x * (blockDim.x >> 5) + (threadIdx.x >> 5);
  if (node >= n) return;
  int lane = threadIdx.x & 31;
  unsigned beg = (node == 0) ? 0u : rs[node - 1];
  unsigned end = rs[node];
  const float dd = dinv[node];
  const float4* __restrict__ h4 = (const float4*)h;

  // self-loop term: h[node] * dinv[node]^2
  float4 hv = h4[(size_t)node * 32 + lane];
  float d2 = dd * dd;
  float4 acc;
  acc.x = hv.x * d2; acc.y = hv.y * d2; acc.z = hv.z * d2; acc.w = hv.w * d2;

  unsigned p = beg;
  for (; p + 2 <= end; p += 2) {                 // 2-way unroll for load overlap
    unsigned s0 = esrc[p], s1 = esrc[p + 1];
    float n0 = dinv[s0] * dd, n1 = dinv[s1] * dd;
    float4 v0 = h4[(size_t)s0 * 32 + lane];
    float4 v1 = h4[(size_t)s1 * 32 + lane];
    acc.x = fmaf(v0.x, n0, acc.x); acc.y = fmaf(v0.y, n0, acc.y);
    acc.z = fmaf(v0.z, n0, acc.z); acc.w = fmaf(v0.w, n0, acc.w);
    acc.x = fmaf(v1.x, n1, acc.x); acc.y = fmaf(v1.y, n1, acc.y);
    acc.z = fmaf(v1.z, n1, acc.z); acc.w = fmaf(v1.w, n1, acc.w);
  }
  if (p < end) {
    unsigned s0 = esrc[p];
    float n0 = dinv[s0] * dd;
    float4 v0 = h4[(size_t)s0 * 32 + lane];
    acc.x = fmaf(v0.x, n0, acc.x); acc.y = fmaf(v0.y, n0, acc.y);
    acc.z = fmaf(v0.z, n0, acc.z); acc.w = fmaf(v0.w, n0, acc.w);
  }

  float4 bv = ((const float4*)bias)[lane];
  acc.x = fmaxf(acc.x + bv.x, 0.0f);
  acc.y = fmaxf(acc.y + bv.y, 0.0f);
  acc.z = fmaxf(acc.z + bv.z, 0.0f);
  acc.w = fmaxf(acc.w + bv.w, 0.0f);
  ((float4*)out)[(size_t)node * 32 + lane] = acc;
}

extern "C" void kernel_launch(void* const* d_in, const int* in_sizes, int n_in,
                              void* d_out, int out_size, void* d_ws, size_t ws_size,
                              hipStream_t stream) {
  const float* x      = (const float*)d_in[0];
  const long long* ei = (const long long*)d_in[1];  // int64 per reference
  const float* W1     = (const float*)d_in[2];
  const float* b1     = (const float*)d_in[3];
  const float* W2     = (const float*)d_in[4];
  const float* b2     = (const float*)d_in[5];
  float* out = (float*)d_out;

  const int n = in_sizes[0] / 128;
  const int e = in_sizes[1] / 2;

  // ---- workspace carve (aligned 512B regions) ----
  char* w = (char*)d_ws;
  auto carve = [&](size_t bytes) { char* p = w; w += (bytes + 511) & ~(size_t)511; return p; };
  unsigned* counts = (unsigned*)carve((size_t)n * 4);
  unsigned* rs     = (unsigned*)carve((size_t)n * 4);
  unsigned* cur    = (unsigned*)carve((size_t)n * 4);
  unsigned* bsum   = (unsigned*)carve(512 * 4);
  float*    dinv   = (float*)carve((size_t)n * 4);
  unsigned* esrc   = (unsigned*)carve((size_t)e * 4);
  float*    h      = (float*)carve((size_t)n * 128 * 4);

  const int nb_n    = (n + THREADS - 1) / THREADS;
  const int nb_e    = (e + THREADS - 1) / THREADS;
  const int nb_scan = (n + 255) / 256;            // 391 <= 512
  const int gemm_blocks = (n + 127) / 128;
  const int agg_blocks  = (n + 7) / 8;            // 8 waves/block, 1 node/wave

  // ---- CSR build + normalization (int atomics only) ----
  k_zero_u32<<<nb_n, THREADS, 0, stream>>>(counts, n);
  k_count<<<nb_e, THREADS, 0, stream>>>(ei + e, counts, e);
  k_dinv<<<nb_n, THREADS, 0, stream>>>(counts, dinv, n);
  k_scan1<<<nb_scan, 256, 0, stream>>>(counts, rs, bsum, n);
  k_scan2<<<1, 512, 0, stream>>>(bsum, nb_scan);
  k_scan3<<<nb_scan, 256, 0, stream>>>(rs, bsum, n);
  k_cursor<<<nb_n, THREADS, 0, stream>>>(rs, cur, n);
  k_fill_edges<<<nb_e, THREADS, 0, stream>>>(ei, cur, esrc, e);

  // ---- layer 1: WMMA projection, then fused gather-aggregate (+bias, ReLU) ----
  k_gemm128_wmma<<<gemm_blocks, THREADS, 64 * 1024, stream>>>(x, W1, h, n);
  k_aggregate<<<agg_blocks, THREADS, 0, stream>>>(rs, esrc, dinv, h, b1, out, n);

  // ---- layer 2 (gemm consumes d_out fully before aggregate overwrites it) ----
  k_gemm128_wmma<<<gemm_blocks, THREADS, 64 * 1024, stream>>>(out, W2, h, n);
  k_aggregate<<<agg_blocks, THREADS, 0, stream>>>(rs, esrc, dinv, h, b2, out, n);
}